// GraphSAGE_63814624084110
// MI455X (gfx1250) — compile-verified
//
#include <hip/hip_runtime.h>
#include <hip/hip_bf16.h>

#define NODES   100000
#define HID     128
#define LAYERS  3
#define EDGES   1600000
#define LN_EPS  1e-5f
#define WPAD    130   // LDS row stride in floats (128 + 2 pad -> bank-conflict-free)

typedef __attribute__((ext_vector_type(2))) float    v2f;
typedef __attribute__((ext_vector_type(8))) float    v8f;
typedef __attribute__((ext_vector_type(4))) unsigned u32x4;
typedef __attribute__((ext_vector_type(8))) unsigned u32x8;

// D = A(16x4 f32) * B(4x16 f32) + C(16x16 f32), full fp32 precision on the WMMA pipe.
__device__ __forceinline__ v8f wmma_f32(v2f a, v2f b, v8f c) {
    return __builtin_amdgcn_wmma_f32_16x16x4_f32(
        /*neg_a=*/false, a, /*neg_b=*/false, b,
        /*c_mod=*/(short)0, c, /*reuse_a=*/false, /*reuse_b=*/false);
}

// TDM: DMA a 128x128 f32 tile (row stride gstride_elems in global) into LDS at
// lds_byte_off, with hardware LDS padding of 2 DWORDs every 128 DWORDs so the
// staged tile has a 130-float row stride (bank-conflict-free ds_load_b64).
// D# per CDNA5 ISA ch.8: group0 = {count/lds_addr/global_addr/type=2},
// group1 = {data_size=4B, pad cfg, dims/tile/stride}. 2D tensor -> 2-operand form
// (VADDR2/VADDR3 = NULL).
__device__ __forceinline__ void tdm_load_tile128(unsigned lds_byte_off,
                                                 const float* gsrc,
                                                 unsigned gstride_elems)
{
    unsigned long long ga = (unsigned long long)(size_t)gsrc;
    u32x4 g0;
    g0[0] = 1u;                                   // count=1 valid user descriptor
    g0[1] = lds_byte_off;                         // lds_addr
    g0[2] = (unsigned)ga;                         // global_addr[31:0]
    g0[3] = (unsigned)(ga >> 32) | 0x80000000u;   // global_addr[56:32] | type=2
    u32x8 g1;
    g1[0] = (2u << 16)      // data_size = 4 bytes
          | (1u << 20)      // pad_enable
          | (6u << 22)      // pad_interval: every 128 DWORDs
          | (1u << 25);     // pad_amount: 2 DWORDs
    g1[1] = (128u << 16);   // tensor_dim0 = 128
    g1[2] = (128u << 16);   // tensor_dim1 = 128
    g1[3] = (128u << 16);   // tile_dim0   = 128
    g1[4] = 128u;           // tile_dim1   = 128, tile_dim2 = 0
    g1[5] = gstride_elems;  // tensor_dim0_stride[31:0]
    g1[6] = 0u;
    g1[7] = 0u;
    asm volatile("tensor_load_to_lds %0, %1" :: "s"(g0), "s"(g1) : "memory");
}

__global__ void zero_f32(float* __restrict__ p, long n) {
    long i = (long)blockIdx.x * blockDim.x + threadIdx.x;
    long stride = (long)gridDim.x * blockDim.x;
    for (; i < n; i += stride) p[i] = 0.0f;
}

__global__ void degree_kernel(const int* __restrict__ dst, float* __restrict__ deg) {
    int i = blockIdx.x * blockDim.x + threadIdx.x;
    int stride = gridDim.x * blockDim.x;
    for (; i < EDGES; i += stride)
        atomicAdd(&deg[dst[i]], 1.0f);
}

__global__ void invdeg_kernel(float* __restrict__ deg) {
    int i = blockIdx.x * blockDim.x + threadIdx.x;
    if (i < NODES) deg[i] = 1.0f / fmaxf(deg[i], 1.0f);
}

// One wave32 per edge: each lane moves float4 (whole 512B row per wave, coalesced),
// scatter via global_atomic_add_f32 into the L2-resident agg buffer.
__global__ void __launch_bounds__(256)
scatter_kernel(const float* __restrict__ x,
               const int* __restrict__ src,
               const int* __restrict__ dst,
               float* __restrict__ agg) {
    const int wave = threadIdx.x >> 5;
    const int lane = threadIdx.x & 31;
    const int e = blockIdx.x * (blockDim.x >> 5) + wave;
    if (e >= EDGES) return;
    const int s = src[e];
    const int d = dst[e];
    const float4 v = reinterpret_cast<const float4*>(x + (size_t)s * HID)[lane];
    float* ar = agg + (size_t)d * HID + lane * 4;
    atomicAdd(ar + 0, v.x);
    atomicAdd(ar + 1, v.y);
    atomicAdd(ar + 2, v.z);
    atomicAdd(ar + 3, v.w);
}

// One wave per 16-row block: 8 WMMA C-tiles cover all 128 output cols.
// Weights (Wl, Wr) are TDM-staged once per workgroup into padded LDS; the
// K-loop reads B fragments with conflict-free ds_load_b64 and only streams
// A rows from global. Fuses bias + LayerNorm + ReLU + residual.
__global__ void __launch_bounds__(256)
sage_layer_kernel(const float* __restrict__ agg,
                  const float* __restrict__ inv_deg,
                  const float* __restrict__ x_in,
                  const float* __restrict__ Wl, const float* __restrict__ bl,
                  const float* __restrict__ Wr,
                  const float* __restrict__ gamma, const float* __restrict__ beta,
                  float* __restrict__ x_out, int add_residual)
{
    __shared__ float smem[2 * HID * WPAD];        // 133,120 B: Wl | Wr (padded)

    const int wave   = threadIdx.x >> 5;
    const int lane   = threadIdx.x & 31;
    const int nblk   = NODES / 16;
    int rowblk = blockIdx.x * (blockDim.x >> 5) + wave;
    const bool active = rowblk < nblk;            // wave-uniform
    if (!active) rowblk = nblk - 1;               // clamp: safe reads, no stores

    if (threadIdx.x == 0) {
        const unsigned base = (unsigned)(size_t)(const void*)smem; // LDS byte off
        tdm_load_tile128(base,                  Wl, HID);
        tdm_load_tile128(base + HID * WPAD * 4, Wr, HID);
        __builtin_amdgcn_s_wait_tensorcnt(0);
    }
    __syncthreads();

    const int hi = lane >> 4;           // K-half select (ISA A 16x4 layout)
    const int lo = lane & 15;           // A row / B col / C col within tile
    const int row0 = rowblk * 16;
    const int arow = row0 + lo;
    const float sc = inv_deg[arow];

    v8f acc[8];
    #pragma unroll
    for (int t = 0; t < 8; ++t)
        #pragma unroll
        for (int v = 0; v < 8; ++v) acc[t][v] = 0.0f;

    const float* aggr = agg  + (size_t)arow * HID;
    const float* xr   = x_in + (size_t)arow * HID;
    const float* wl_s = smem;
    const float* wr_s = smem + HID * WPAD;

    #pragma unroll 4
    for (int ks = 0; ks < HID / 4; ++ks) {
        const int k = ks * 4 + hi * 2;
        v2f a1, a2;
        a1.x = aggr[k] * sc;  a1.y = aggr[k + 1] * sc;   // fold mean-scale into A frag
        a2.x = xr[k];         a2.y = xr[k + 1];
        #pragma unroll
        for (int t = 0; t < 8; ++t) {
            const int col = t * 16 + lo;
            const v2f b1 = *reinterpret_cast<const v2f*>(wl_s + col * WPAD + k);
            const v2f b2 = *reinterpret_cast<const v2f*>(wr_s + col * WPAD + k);
            acc[t] = wmma_f32(a1, b1, acc[t]);
            acc[t] = wmma_f32(a2, b2, acc[t]);
        }
    }

    // + bias (bias depends only on col = t*16+lo)
    #pragma unroll
    for (int t = 0; t < 8; ++t) {
        const float b = bl[t * 16 + lo];
        #pragma unroll
        for (int v = 0; v < 8; ++v) acc[t][v] += b;
    }

    // LayerNorm per row. C layout: lanes 0-15 hold rows v, lanes 16-31 rows v+8,
    // so each 16-lane half reduces its own rows via width-16 shfl_xor.
    #pragma unroll
    for (int v = 0; v < 8; ++v) {
        float s = 0.0f;
        #pragma unroll
        for (int t = 0; t < 8; ++t) s += acc[t][v];
        s += __shfl_xor(s, 8, 16);
        s += __shfl_xor(s, 4, 16);
        s += __shfl_xor(s, 2, 16);
        s += __shfl_xor(s, 1, 16);
        const float mu = s * (1.0f / HID);

        float q = 0.0f;
        #pragma unroll
        for (int t = 0; t < 8; ++t) { const float d = acc[t][v] - mu; q += d * d; }
        q += __shfl_xor(q, 8, 16);
        q += __shfl_xor(q, 4, 16);
        q += __shfl_xor(q, 2, 16);
        q += __shfl_xor(q, 1, 16);
        const float rs = rsqrtf(q * (1.0f / HID) + LN_EPS);

        const int m = v + 8 * hi;
        const size_t rowoff = (size_t)(row0 + m) * HID;
        #pragma unroll
        for (int t = 0; t < 8; ++t) {
            const int col = t * 16 + lo;
            float val = (acc[t][v] - mu) * rs * gamma[col] + beta[col];
            val = fmaxf(val, 0.0f);
            if (add_residual) val += x_in[rowoff + col];
            if (active) x_out[rowoff + col] = val;
        }
    }
}

// Jumping-knowledge: out = [h0|h1|h2|h3] @ jkW^T + jkb.  jkW is 256 KB, so it is
// TDM-staged as four 64 KB strided slices (tensor_dim0_stride = 512), one per
// concat block, in a barrier-pipelined stage loop; accumulators live in VGPRs.
__global__ void __launch_bounds__(256)
jk_kernel(const float* __restrict__ h0, const float* __restrict__ h1,
          const float* __restrict__ h2, const float* __restrict__ h3,
          const float* __restrict__ jkW, const float* __restrict__ jkb,
          float* __restrict__ out)
{
    __shared__ float smem[HID * WPAD];            // 66,560 B: one jkW slice (padded)

    const int wave   = threadIdx.x >> 5;
    const int lane   = threadIdx.x & 31;
    const int nblk   = NODES / 16;
    int rowblk = blockIdx.x * (blockDim.x >> 5) + wave;
    const bool active = rowblk < nblk;
    if (!active) rowblk = nblk - 1;

    const int hi = lane >> 4;
    const int lo = lane & 15;
    const int row0 = rowblk * 16;
    const int arow = row0 + lo;

    v8f acc[8];
    #pragma unroll
    for (int t = 0; t < 8; ++t)
        #pragma unroll
        for (int v = 0; v < 8; ++v) acc[t][v] = 0.0f;

    const float* srcs[4] = { h0, h1, h2, h3 };
    for (int s = 0; s < 4; ++s) {
        __syncthreads();                          // prior-stage reads done (WAR)
        if (threadIdx.x == 0) {
            const unsigned base = (unsigned)(size_t)(const void*)smem;
            tdm_load_tile128(base, jkW + s * HID, 4 * HID);  // strided slice
            __builtin_amdgcn_s_wait_tensorcnt(0);
        }
        __syncthreads();

        const float* hr = srcs[s] + (size_t)arow * HID;
        #pragma unroll 2
        for (int ks = 0; ks < HID / 4; ++ks) {
            const int k = ks * 4 + hi * 2;
            v2f a; a.x = hr[k]; a.y = hr[k + 1];
            #pragma unroll
            for (int t = 0; t < 8; ++t) {
                const int col = t * 16 + lo;
                const v2f b = *reinterpret_cast<const v2f*>(smem + col * WPAD + k);
                acc[t] = wmma_f32(a, b, acc[t]);
            }
        }
    }

    #pragma unroll
    for (int v = 0; v < 8; ++v) {
        const int m = v + 8 * hi;
        const size_t rowoff = (size_t)(row0 + m) * HID;
        #pragma unroll
        for (int t = 0; t < 8; ++t) {
            const int col = t * 16 + lo;
            if (active) out[rowoff + col] = acc[t][v] + jkb[col];
        }
    }
}

extern "C" void kernel_launch(void* const* d_in, const int* in_sizes, int n_in,
                              void* d_out, int out_size, void* d_ws, size_t ws_size,
                              hipStream_t stream)
{
    const float* emb = (const float*)d_in[0];
    const float* Wl  = (const float*)d_in[1];
    const float* bl  = (const float*)d_in[2];
    const float* Wr  = (const float*)d_in[3];
    const float* gam = (const float*)d_in[4];
    const float* bet = (const float*)d_in[5];
    const float* jkW = (const float*)d_in[6];
    const float* jkb = (const float*)d_in[7];
    const int*  eidx = (const int*)d_in[8];
    const int* src = eidx;            // edge_index row 0
    const int* dst = eidx + EDGES;    // edge_index row 1
    float* out = (float*)d_out;

    // workspace carve-out (256B aligned): inv_deg[N], agg[N*H], h1..h3[N*H]
    char* ws = (char*)d_ws;
    size_t o = 0;
    auto carve = [&](size_t bytes) {
        void* p = ws + o;
        o += (bytes + 255) & ~(size_t)255;
        return p;
    };
    float* inv_deg = (float*)carve((size_t)NODES * 4);
    float* aggbuf  = (float*)carve((size_t)NODES * HID * 4);
    float* h1      = (float*)carve((size_t)NODES * HID * 4);
    float* h2      = (float*)carve((size_t)NODES * HID * 4);
    float* h3      = (float*)carve((size_t)NODES * HID * 4);
    (void)ws_size; (void)in_sizes; (void)n_in; (void)out_size;

    // degrees -> inverse degrees (in place)
    zero_f32<<<1024, 256, 0, stream>>>(inv_deg, (long)NODES);
    degree_kernel<<<2048, 256, 0, stream>>>(dst, inv_deg);
    invdeg_kernel<<<(NODES + 255) / 256, 256, 0, stream>>>(inv_deg);

    const int nblk      = NODES / 16;           // 6250 16-row blocks
    const int gemm_grid = (nblk + 7) / 8;       // 8 waves / block

    const float* xin = emb;
    float* houts[LAYERS] = { h1, h2, h3 };
    for (int i = 0; i < LAYERS; ++i) {
        zero_f32<<<4096, 256, 0, stream>>>(aggbuf, (long)NODES * HID);
        scatter_kernel<<<EDGES / 8, 256, 0, stream>>>(xin, src, dst, aggbuf);
        sage_layer_kernel<<<gemm_grid, 256, 0, stream>>>(
            aggbuf, inv_deg, xin,
            Wl + (size_t)i * HID * HID, bl + (size_t)i * HID,
            Wr + (size_t)i * HID * HID,
            gam + (size_t)i * HID, bet + (size_t)i * HID,
            houts[i], i > 0 ? 1 : 0);
        xin = houts[i];
    }

    jk_kernel<<<gemm_grid, 256, 0, stream>>>(emb, h1, h2, h3, jkW, jkb, out);
}